// SA_70858370449447
// MI455X (gfx1250) — compile-verified
//
#include <hip/hip_runtime.h>
#include <math.h>

typedef __attribute__((ext_vector_type(2))) float v2f;
typedef __attribute__((ext_vector_type(8))) float v8f;

#define NLAYERS 2
#define KHEADS  8
#define HDIM    64
#define DDIM    1024
#define BATCH   4
#define LSEQ    1024
#define HKDIM   512            // HDIM*KHEADS
#define NROWS   (BATCH*LSEQ)   // 4096
#define DHEAD   128            // DDIM/KHEADS

__device__ __forceinline__ v8f wmma_f32(v2f a, v2f b, v8f c) {
  // D = A(16x4 f32) * B(4x16 f32) + C(16x16 f32)
  return __builtin_amdgcn_wmma_f32_16x16x4_f32(false, a, false, b, (short)0, c,
                                               false, false);
}

// ---------------------------------------------------------------------------
// out[row, col] = relu(in[row,:] @ W[:,col] + bias[col])
// in: [4096,1024], W: [1024,512], out: [4096,512]
// Block: 128 threads (4 waves). Block tile: 16 rows x 64 cols, 1 wave = 16x16.
// ---------------------------------------------------------------------------
__global__ void proj_relu_kernel(const float* __restrict__ in,
                                 const float* __restrict__ W,
                                 const float* __restrict__ bias,
                                 float* __restrict__ outb) {
  extern __shared__ float smem[];
  float* sA = smem;           // 16*64 = 1024 floats
  float* sB = smem + 1024;    // 64*64 = 4096 floats

  const int tid  = threadIdx.x;
  const int lane = tid & 31;
  const int wave = tid >> 5;
  const int half = lane >> 4;   // 0: lanes 0-15, 1: lanes 16-31
  const int ln   = lane & 15;
  const int row0 = blockIdx.x * 16;
  const int col0 = blockIdx.y * 64;

  v8f acc = {0.f, 0.f, 0.f, 0.f, 0.f, 0.f, 0.f, 0.f};

  for (int kk = 0; kk < DDIM; kk += 64) {
    for (int i = tid; i < 16 * 64; i += 128) {
      int r = i >> 6, c = i & 63;
      sA[i] = in[(size_t)(row0 + r) * DDIM + kk + c];
    }
    for (int i = tid; i < 64 * 64; i += 128) {
      int r = i >> 6, c = i & 63;
      sB[i] = W[(size_t)(kk + r) * HKDIM + col0 + c];
    }
    __syncthreads();
#pragma unroll
    for (int t = 0; t < 16; ++t) {
      const int kb = t * 4 + half * 2;
      v2f a = *(const v2f*)&sA[ln * 64 + kb];
      v2f b;
      b.x = sB[kb * 64 + wave * 16 + ln];
      b.y = sB[(kb + 1) * 64 + wave * 16 + ln];
      acc = wmma_f32(a, b, acc);
    }
    __syncthreads();
  }

  const int coln = col0 + wave * 16 + ln;
  const float bv = bias[coln];
#pragma unroll
  for (int j = 0; j < 8; ++j) {
    const int m = j + half * 8;
    float v = acc[j] + bv;
    v = v > 0.f ? v : 0.f;
    outb[(size_t)(row0 + m) * HKDIM + coln] = v;
  }
}

// ---------------------------------------------------------------------------
// Fused attention for one (batch b, head kh, 16-row q tile):
//   S = q k^T * 0.125, strict-lower causal mask (-1e30 where s >= qrow)
//   att = softmax(S); write att to graphs; cur = att @ v; out = inp + cur
// Block: 128 threads. Dynamic LDS: att strip(16384) + stage(8192) + red(160).
// ---------------------------------------------------------------------------
__global__ void attn_kernel(const float* __restrict__ inp,
                            const float* __restrict__ qbuf,
                            const float* __restrict__ kbuf,
                            float* __restrict__ graphs,
                            float* __restrict__ outp) {
  extern __shared__ float smem[];
  float* attbuf = smem;                   // 16 x 1024
  float* stage  = smem + 16384;           // 8192 floats (v chunk / q+k staging)
  float* red    = smem + 16384 + 8192;    // 128 partials + 16 row values

  const int tid  = threadIdx.x;
  const int lane = tid & 31;
  const int wave = tid >> 5;
  const int half = lane >> 4;
  const int ln   = lane & 15;
  const int qt   = blockIdx.x;            // q tile 0..63
  const int kh   = blockIdx.y;            // head 0..7
  const int b    = blockIdx.z;            // batch 0..3
  const int row0 = qt * 16;

  float* sQ = stage;                      // 16*64
  float* sK = stage + 1024 + wave * 1024; // 16*64 per wave

  // ---- load q tile -------------------------------------------------------
  for (int i = tid; i < 1024; i += 128) {
    int r = i >> 6, c = i & 63;
    sQ[i] = qbuf[(size_t)(b * LSEQ + row0 + r) * HKDIM + kh * HDIM + c];
  }
  __syncthreads();

  // ---- scores: each wave handles s-tiles j = wave, wave+4, ... -----------
  for (int j = wave; j < 64; j += 4) {
    for (int i = lane; i < 1024; i += 32) {
      int r = i >> 6, c = i & 63;
      sK[i] = kbuf[(size_t)(b * LSEQ + j * 16 + r) * HKDIM + kh * HDIM + c];
    }
    v8f acc = {0.f, 0.f, 0.f, 0.f, 0.f, 0.f, 0.f, 0.f};
#pragma unroll
    for (int t = 0; t < 16; ++t) {
      const int kb = t * 4 + half * 2;
      v2f a = *(const v2f*)&sQ[ln * 64 + kb];
      v2f bb;
      bb.x = sK[kb * 64 + ln];
      bb.y = sK[(kb + 1) * 64 + ln];
      acc = wmma_f32(a, bb, acc);
    }
#pragma unroll
    for (int jj = 0; jj < 8; ++jj) {
      const int m  = jj + half * 8;
      const int sg = j * 16 + ln;
      float v = acc[jj] * 0.125f;           // 1/sqrt(64)
      if (sg >= row0 + m) v -= 1e30f;       // strict lower-triangular mask
      attbuf[m * 1024 + sg] = v;
    }
  }
  __syncthreads();

  // ---- softmax over s (rows of att strip); 8 threads per row -------------
  const int srow = tid >> 3;   // 0..15
  const int seg  = tid & 7;    // 0..7
  float* myrow = attbuf + srow * 1024 + seg * 128;

  float mx = -3.0e38f;
  for (int c = 0; c < 128; ++c) mx = fmaxf(mx, myrow[c]);
  red[tid] = mx;
  __syncthreads();
  if (seg == 0) {
    float m2 = red[tid];
    for (int s2 = 1; s2 < 8; ++s2) m2 = fmaxf(m2, red[tid + s2]);
    red[128 + srow] = m2;
  }
  __syncthreads();
  const float rmax = red[128 + srow];
  float psum = 0.f;
  for (int c = 0; c < 128; ++c) {
    float e = expf(myrow[c] - rmax);
    myrow[c] = e;
    psum += e;
  }
  red[tid] = psum;
  __syncthreads();
  if (seg == 0) {
    float sv = 0.f;
    for (int s2 = 0; s2 < 8; ++s2) sv += red[tid + s2];
    red[128 + srow] = sv;
  }
  __syncthreads();
  const float rinv = 1.0f / red[128 + srow];
  for (int c = 0; c < 128; ++c) myrow[c] *= rinv;
  __syncthreads();

  // ---- coalesced write of normalized att to graphs ------------------------
  float* g = graphs + ((size_t)(b * KHEADS + kh) * LSEQ + row0) * LSEQ;
  for (int i = tid; i < 4096; i += 128) {       // 16*1024 floats as float4
    int r  = i >> 8;                            // 256 float4 per row
    int c4 = i & 255;
    float4 v4 = *(const float4*)&attbuf[r * 1024 + c4 * 4];
    *(float4*)&g[(size_t)r * LSEQ + c4 * 4] = v4;
  }

  // ---- AV: cur[16,128] = att[16,1024] @ v[1024,128] -----------------------
  v8f acc0 = {0.f, 0.f, 0.f, 0.f, 0.f, 0.f, 0.f, 0.f};
  v8f acc1 = {0.f, 0.f, 0.f, 0.f, 0.f, 0.f, 0.f, 0.f};
  const int d0 = wave * 32;   // this wave: d cols [d0, d0+32)

  for (int s0 = 0; s0 < LSEQ; s0 += 64) {
    __syncthreads();          // previous chunk fully consumed
    for (int i = tid; i < 2048; i += 128) {     // 64*128 floats as float4
      int r  = i >> 5;                          // 32 float4 per row
      int c4 = i & 31;
      *(float4*)&stage[r * 128 + c4 * 4] =
          *(const float4*)&inp[(size_t)(b * LSEQ + s0 + r) * DDIM +
                               kh * DHEAD + c4 * 4];
    }
    __syncthreads();
#pragma unroll
    for (int t = 0; t < 16; ++t) {
      const int sr = t * 4 + half * 2;
      v2f a = *(const v2f*)&attbuf[ln * 1024 + s0 + sr];
      v2f b0, b1;
      b0.x = stage[sr * 128 + d0 + ln];
      b0.y = stage[(sr + 1) * 128 + d0 + ln];
      acc0 = wmma_f32(a, b0, acc0);
      b1.x = stage[sr * 128 + d0 + 16 + ln];
      b1.y = stage[(sr + 1) * 128 + d0 + 16 + ln];
      acc1 = wmma_f32(a, b1, acc1);
    }
  }

  // ---- residual add + store ----------------------------------------------
#pragma unroll
  for (int jj = 0; jj < 8; ++jj) {
    const int m = jj + half * 8;
    const size_t base = (size_t)(b * LSEQ + row0 + m) * DDIM + kh * DHEAD;
    const int c0 = d0 + ln, c1 = d0 + 16 + ln;
    outp[base + c0] = inp[base + c0] + acc0[jj];
    outp[base + c1] = inp[base + c1] + acc1[jj];
  }
}

// ---------------------------------------------------------------------------
extern "C" void kernel_launch(void* const* d_in, const int* in_sizes, int n_in,
                              void* d_out, int out_size, void* d_ws,
                              size_t ws_size, hipStream_t stream) {
  const float* x  = (const float*)d_in[0];
  const float* Wq = (const float*)d_in[1];
  const float* bq = (const float*)d_in[2];
  const float* Wk = (const float*)d_in[3];
  const float* bk = (const float*)d_in[4];

  float* graphs = (float*)d_out;
  const size_t GPL = (size_t)BATCH * KHEADS * LSEQ * LSEQ;  // graphs per layer
  float* outs = graphs + (size_t)NLAYERS * GPL;             // outputs section

  float* qbuf = (float*)d_ws;                 // [4096, 512]
  float* kbuf = qbuf + (size_t)NROWS * HKDIM; // [4096, 512]

  const size_t projLds = (size_t)(1024 + 4096) * sizeof(float);          // 20 KB
  const size_t attnLds = (size_t)(16384 + 8192 + 160) * sizeof(float);   // ~97 KB

  for (int layer = 0; layer < NLAYERS; ++layer) {
    const float* in =
        (layer == 0) ? x : (outs + (size_t)(layer - 1) * BATCH * LSEQ * DDIM);
    float* outL = outs + (size_t)layer * BATCH * LSEQ * DDIM;

    proj_relu_kernel<<<dim3(NROWS / 16, HKDIM / 64), 128, projLds, stream>>>(
        in, Wq + (size_t)layer * DDIM * HKDIM, bq + (size_t)layer * HKDIM, qbuf);
    proj_relu_kernel<<<dim3(NROWS / 16, HKDIM / 64), 128, projLds, stream>>>(
        in, Wk + (size_t)layer * DDIM * HKDIM, bk + (size_t)layer * HKDIM, kbuf);
    attn_kernel<<<dim3(LSEQ / 16, KHEADS, BATCH), 128, attnLds, stream>>>(
        in, qbuf, kbuf, graphs + (size_t)layer * GPL, outL);
  }
}